// QuantumKANRegressor_26688926777596
// MI455X (gfx1250) — compile-verified
//
#include <hip/hip_runtime.h>
#include <hip/hip_bf16.h>

typedef float v2f __attribute__((ext_vector_type(2)));
typedef float v4f __attribute__((ext_vector_type(4)));
typedef float v8f __attribute__((ext_vector_type(8)));

#define F_DIM 256
#define ROWS_PER_BLOCK 128   // 8 waves * 16 rows

static __device__ __forceinline__ float fast_tanh(float x) {
#if __has_builtin(__builtin_amdgcn_tanhf)
    return __builtin_amdgcn_tanhf(x);          // v_tanh_f32 (gfx1250 TRANS op)
#elif __has_builtin(__builtin_amdgcn_tanh_f32)
    return __builtin_amdgcn_tanh_f32(x);
#else
    return tanhf(x);
#endif
}

// One wave handles 16 batch rows. Lane L -> row (L>>1), feature half (L&1).
// Each lane streams 128 of the 256 features of its row via float4 loads,
// keeping two accumulators (even/odd iteration) -> 4 partials per row across
// 2 lanes. ds_bpermute transposes those into the WMMA A-matrix layout
// (16x4 f32: lane r<16 holds K=0,1 ; lane 16+r holds K=2,3), and a single
// v_wmma_f32_16x16x4_f32 against an all-ones B reduces them to 16 per-row
// sums in D's N=0 column (all columns identical).
__global__ __launch_bounds__(256) void qkan_fused_kernel(
    const float* __restrict__ X,        // (B, 256)
    const float* __restrict__ lcu_w,    // (256, 5)
    const float* __restrict__ sum_w,    // (256, 5)
    const float* __restrict__ kan_c,    // (1, 256, 6)
    float* __restrict__ out,            // (B, 1)
    int B)
{
    // Per-feature constants packed as 3 float4 (one 48B record per feature):
    //  q0 = { c0, c1, c2, c3 }          c_k = lcu_w/sum|lcu_w| * sum_w  (T_1..T_5 weights)
    //  q1 = { c4, k0, k1, k2 }          k_j = kan coefficients (T_0..T_5 weights)
    //  q2 = { k3, k4, k5, 0  }
    __shared__ v4f sc4[F_DIM][3];

    const int tid = threadIdx.x;

    // ---- per-feature constant precompute (256 threads == 256 features) ----
    {
        const int f = tid;
        float w0 = lcu_w[f * 5 + 0];
        float w1 = lcu_w[f * 5 + 1];
        float w2 = lcu_w[f * 5 + 2];
        float w3 = lcu_w[f * 5 + 3];
        float w4 = lcu_w[f * 5 + 4];
        float inv = 1.0f / (fabsf(w0) + fabsf(w1) + fabsf(w2) + fabsf(w3) + fabsf(w4));
        v4f q0, q1, q2;
        q0.x = w0 * inv * sum_w[f * 5 + 0];
        q0.y = w1 * inv * sum_w[f * 5 + 1];
        q0.z = w2 * inv * sum_w[f * 5 + 2];
        q0.w = w3 * inv * sum_w[f * 5 + 3];
        q1.x = w4 * inv * sum_w[f * 5 + 4];
        q1.y = kan_c[f * 6 + 0];
        q1.z = kan_c[f * 6 + 1];
        q1.w = kan_c[f * 6 + 2];
        q2.x = kan_c[f * 6 + 3];
        q2.y = kan_c[f * 6 + 4];
        q2.z = kan_c[f * 6 + 5];
        q2.w = 0.0f;
        sc4[f][0] = q0;
        sc4[f][1] = q1;
        sc4[f][2] = q2;
    }
    __syncthreads();

    const int lane = tid & 31;
    const int wave = tid >> 5;
    const int rowBase = blockIdx.x * ROWS_PER_BLOCK + wave * 16;

    int row = rowBase + (lane >> 1);
    if (row >= B) row = B - 1;                 // clamp (B % 128 == 0 in practice)
    const float* __restrict__ xrow = X + (size_t)row * F_DIM;
    const int fhalf = (lane & 1) * 4;          // which float4 of each 8-feature chunk

    float acc0 = 0.0f, acc1 = 0.0f;

#pragma unroll 4
    for (int i = 0; i < 32; ++i) {
        const int f0 = i * 8 + fhalf;
        if ((i & 3) == 0) {
            // keep the HBM stream 512B (four 128B lines) ahead (speculative, OOB-safe)
            __builtin_prefetch(xrow + i * 8 + 128, 0, 0);   // global_prefetch_b8
        }
        v4f xv = *(const v4f*)(xrow + f0);     // global_load_b128, 16B contiguous
        float g = 0.0f;
#pragma unroll
        for (int j = 0; j < 4; ++j) {
            const float x = xv[j];
            const v4f q0 = sc4[f0 + j][0];     // ds_load_b128 x3
            const v4f q1 = sc4[f0 + j][1];
            const v4f q2 = sc4[f0 + j][2];
            const float x2 = x + x;
            // T_1..T_5(x) via recurrence (exact for x in [-1,1]; X in [0,1))
            const float t1 = x;
            const float t2 = __builtin_fmaf(x2, x, -1.0f);
            const float t3 = __builtin_fmaf(x2, t2, -t1);
            const float t4 = __builtin_fmaf(x2, t3, -t2);
            const float t5 = __builtin_fmaf(x2, t4, -t3);
            float feat = t1 * q0.x;
            feat = __builtin_fmaf(t2, q0.y, feat);
            feat = __builtin_fmaf(t3, q0.z, feat);
            feat = __builtin_fmaf(t4, q0.w, feat);
            feat = __builtin_fmaf(t5, q1.x, feat);
            const float z = fast_tanh(feat);
            const float z2 = z + z;
            // T_0..T_5(z)
            const float u1 = z;
            const float u2 = __builtin_fmaf(z2, z, -1.0f);
            const float u3 = __builtin_fmaf(z2, u2, -u1);
            const float u4 = __builtin_fmaf(z2, u3, -u2);
            const float u5 = __builtin_fmaf(z2, u4, -u3);
            float gg = q1.y;                    // T_0 == 1
            gg = __builtin_fmaf(u1, q1.z, gg);
            gg = __builtin_fmaf(u2, q1.w, gg);
            gg = __builtin_fmaf(u3, q2.x, gg);
            gg = __builtin_fmaf(u4, q2.y, gg);
            gg = __builtin_fmaf(u5, q2.z, gg);
            g += gg;
        }
        if (i & 1) acc1 += g; else acc0 += g;
    }

    // ---- transpose 4 partials/row into WMMA A layout via ds_bpermute ----
    // target: lane r (<16) holds P[r][0],P[r][1]; lane 16+r holds P[r][2],P[r][3]
    // source: P[r] = { lane 2r acc0, lane 2r acc1, lane 2r+1 acc0, lane 2r+1 acc1 }
    const int srcLane = (((lane & 15) << 1) | (lane >> 4)) << 2;   // byte index
    v2f A;
    A.x = __int_as_float(__builtin_amdgcn_ds_bpermute(srcLane, __float_as_int(acc0)));
    A.y = __int_as_float(__builtin_amdgcn_ds_bpermute(srcLane, __float_as_int(acc1)));

    v2f Bm; Bm.x = 1.0f; Bm.y = 1.0f;          // all-ones B: D[m][n] = sum_k A[m][k]
    v8f c = {};
    c = __builtin_amdgcn_wmma_f32_16x16x4_f32(
        /*neg_a=*/false, A, /*neg_b=*/false, Bm,
        /*c_mod=*/(short)0, c, /*reuse_a=*/false, /*reuse_b=*/false);

    // D layout (16x16 f32): VGPR m, lanes 0-15 -> M=m ; lanes 16-31 -> M=m+8.
    // Lane 0 holds rows rowBase..rowBase+7, lane 16 holds rowBase+8..rowBase+15.
    if ((lane & 15) == 0) {
        const int base = rowBase + (lane >> 4) * 8;
        if (base + 8 <= B) {
            v4f lo, hi;
            lo.x = c[0]; lo.y = c[1]; lo.z = c[2]; lo.w = c[3];
            hi.x = c[4]; hi.y = c[5]; hi.z = c[6]; hi.w = c[7];
            *(v4f*)(out + base)     = lo;       // global_store_b128
            *(v4f*)(out + base + 4) = hi;       // global_store_b128
        }
    }
}

extern "C" void kernel_launch(void* const* d_in, const int* in_sizes, int n_in,
                              void* d_out, int out_size, void* d_ws, size_t ws_size,
                              hipStream_t stream) {
    const float* X     = (const float*)d_in[0];   // (B, 256)
    const float* lcu_w = (const float*)d_in[1];   // (256, 5)
    const float* sum_w = (const float*)d_in[2];   // (256, 5)
    const float* kan_c = (const float*)d_in[3];   // (1, 256, 6)
    float* out = (float*)d_out;                   // (B, 1)

    const int B = in_sizes[0] / F_DIM;
    const int grid = (B + ROWS_PER_BLOCK - 1) / ROWS_PER_BLOCK;
    qkan_fused_kernel<<<grid, 256, 0, stream>>>(X, lcu_w, sum_w, kan_c, out, B);
}